// LogicDense_47055661695075
// MI455X (gfx1250) — compile-verified
//
#include <hip/hip_runtime.h>
#include <stdint.h>

// LogicDense soft-forward for MI455X (gfx1250).
// Bandwidth-bound gather+polynomial: stage x rows into LDS via gfx1250 async
// global->LDS loads (ASYNCcnt), gather a/b from LDS, 3 FMAs, NT store.

#define IN_DIM   4096
#define ROWS     4          // batch rows staged per block (4 x 16KB = 64KB LDS)
#define NTHREADS 256

typedef int v4i __attribute__((vector_size(16)));
typedef __attribute__((address_space(1))) v4i g_v4i;   // global
typedef __attribute__((address_space(3))) v4i l_v4i;   // LDS

__device__ __forceinline__ void async_copy_b128(const void* gsrc, void* ldst) {
#if __has_builtin(__builtin_amdgcn_global_load_async_to_lds_b128)
  __builtin_amdgcn_global_load_async_to_lds_b128(
      (g_v4i*)(uintptr_t)gsrc,
      (l_v4i*)(uintptr_t)ldst,
      0, 0);
#else
  uint32_t lds_off = (uint32_t)(uintptr_t)ldst;   // low 32 bits of generic shared addr = LDS offset
  asm volatile("global_load_async_to_lds_b128 %0, %1, off"
               :: "v"(lds_off), "v"(gsrc) : "memory");
#endif
}

__device__ __forceinline__ void wait_async0() {
#if __has_builtin(__builtin_amdgcn_s_wait_asynccnt)
  __builtin_amdgcn_s_wait_asynccnt(0);
#else
  asm volatile("s_wait_asynccnt 0" ::: "memory");
#endif
}

// ---------------------------------------------------------------------------
// Kernel 1: coeff[j] = softmax(weight[j, 0:16]) @ T   (T folded into +-1/+-2 sums)
// ---------------------------------------------------------------------------
__global__ __launch_bounds__(256) void logic_coeff_kernel(
    const float* __restrict__ w, float4* __restrict__ coeff, int out_dim) {
  const int j = blockIdx.x * blockDim.x + threadIdx.x;
  if (j >= out_dim) return;

  float v[16];
  float m = -3.402823466e38f;
#pragma unroll
  for (int i = 0; i < 16; ++i) { v[i] = w[j * 16 + i]; m = fmaxf(m, v[i]); }
  float s = 0.f;
#pragma unroll
  for (int i = 0; i < 16; ++i) { v[i] = __expf(v[i] - m); s += v[i]; }
  const float inv = 1.f / s;

  // difflogic T table, columns as +-1/+-2 sums (op order 0..15)
  const float c0 = v[8] + v[9] + v[10] + v[11] + v[12] + v[13] + v[14] + v[15];
  const float c1 = (v[2] + v[3] + v[6] + v[7]) - (v[8] + v[9] + v[12] + v[13]);
  const float c2 = (v[4] + v[5] + v[6] + v[7]) - (v[8] + v[9] + v[10] + v[11]);
  const float c3 = v[1] - v[2] - v[4] - 2.f * v[6] - v[7]
                 + v[8] + 2.f * v[9] + v[11] + v[13] - v[14];

  coeff[j] = make_float4(c0 * inv, c1 * inv, c2 * inv, c3 * inv);
}

// ---------------------------------------------------------------------------
// Kernel 2: main pass. One block stages ROWS rows of x into LDS (async), then
// each thread sweeps output columns: 2 LDS gathers + 3 FMAs + NT store / row.
// ---------------------------------------------------------------------------
__global__ __launch_bounds__(NTHREADS) void logic_dense_kernel(
    const float*  __restrict__ x,
    const float4* __restrict__ coeff,
    const int*    __restrict__ idx0,
    const int*    __restrict__ idx1,
    float*        __restrict__ out,
    int out_dim, int batch) {
  __shared__ float lds[ROWS * IN_DIM];   // 64 KB of the 320 KB WGP pool

  const int tid = threadIdx.x;
  const int r0  = blockIdx.x * ROWS;
  const float* src = x + (size_t)r0 * IN_DIM;

  // Stage ROWS contiguous rows: ROWS*IN_DIM/4 float4 chunks, async to LDS.
#pragma unroll
  for (int i = tid; i < ROWS * IN_DIM / 4; i += NTHREADS)
    async_copy_b128(src + 4 * i, &lds[4 * i]);
  wait_async0();
  __syncthreads();

  for (int j = tid; j < out_dim; j += NTHREADS) {
    const int    pa = idx0[j];
    const int    pb = idx1[j];
    const float4 c  = coeff[j];
#pragma unroll
    for (int rr = 0; rr < ROWS; ++rr) {
      const float a = lds[rr * IN_DIM + pa];
      const float b = lds[rr * IN_DIM + pb];
      const float r = fmaf(c.w, a * b, fmaf(c.z, b, fmaf(c.y, a, c.x)));
      __builtin_nontemporal_store(r, &out[(size_t)(r0 + rr) * out_dim + j]);
    }
  }
}

// ---------------------------------------------------------------------------
extern "C" void kernel_launch(void* const* d_in, const int* in_sizes, int n_in,
                              void* d_out, int out_size, void* d_ws, size_t ws_size,
                              hipStream_t stream) {
  const float* x       = (const float*)d_in[0];
  const int*   indices = (const int*)  d_in[1];   // (2, out_dim)
  const float* weight  = (const float*)d_in[2];   // (out_dim, 16)
  float*       out     = (float*)d_out;

  const int out_dim = in_sizes[2] / 16;           // 8192
  const int batch   = out_size / out_dim;         // 4096

  float4* coeff = (float4*)d_ws;                  // out_dim * 16 B = 128 KB

  logic_coeff_kernel<<<(out_dim + 255) / 256, 256, 0, stream>>>(weight, coeff, out_dim);

  logic_dense_kernel<<<dim3(batch / ROWS), dim3(NTHREADS), 0, stream>>>(
      x, coeff, indices, indices + out_dim, out, out_dim, batch);
}